// MeanShiftClusterer_5385888989492
// MI455X (gfx1250) — compile-verified
//
#include <hip/hip_runtime.h>
#include <hip/hip_bf16.h>
#include <math.h>

// ---------------------------------------------------------------------------
// Mean-shift clustering, N=8192, D=3, H=0.05, 10 iterations.
// All-pairs passes are tiled 16x16 per wave32; the 3-D dot products are done
// with V_WMMA_F32_16X16X4_F32 (D=3 padded to 4, exact f32). exp is emitted as
// a single raw v_exp_f32 via __builtin_amdgcn_exp2f (args <= 0; sub-denormal
// results correctly flush to 0 = negligible kernel weight).
// ---------------------------------------------------------------------------

typedef __attribute__((ext_vector_type(2))) float v2f;
typedef __attribute__((ext_vector_type(8))) float v8f;

#define MS_H2   0.0025f
// C = log2(e) / h^2  : exp(-d2/h2) == exp2(-d2 * C)
#define MS_CC   (1.4426950408889634f / MS_H2)
#define MS_2C   (2.0f * MS_CC)

__device__ __forceinline__ v8f wmma_dot(v2f a, v2f b) {
  v8f c = {0.0f, 0.0f, 0.0f, 0.0f, 0.0f, 0.0f, 0.0f, 0.0f};
  // D(16x16) = A(16x4) x B(4x16) + C, all f32.
  return __builtin_amdgcn_wmma_f32_16x16x4_f32(
      /*neg_a=*/false, a, /*neg_b=*/false, b,
      /*c_mod=*/(short)0, c, /*reuse_a=*/false, /*reuse_b=*/false);
}

__device__ __forceinline__ float fast_exp2(float x) {
  return __builtin_amdgcn_exp2f(x);   // raw v_exp_f32, no denorm fixup
}

// packedQ[j] = {qx, qy, qz, log2(w_j) - |q_j|^2 * C}
// yA[j]      = {qx, qy, qz, |q_j|^2}
__global__ void __launch_bounds__(256) ms_pack(const float* __restrict__ q,
                                               const float* __restrict__ attn,
                                               float4* __restrict__ pq,
                                               float4* __restrict__ ya, int n) {
  int j = blockIdx.x * blockDim.x + threadIdx.x;
  if (j >= n) return;
  float x = q[3 * j + 0], y = q[3 * j + 1], z = q[3 * j + 2];
  float q2 = fmaf(x, x, fmaf(y, y, z * z));
  float w = attn[j];
  pq[j] = make_float4(x, y, z, __log2f(w) - q2 * MS_CC);
  ya[j] = make_float4(x, y, z, q2);
}

// One mean-shift iteration: yout = (K @ q) / rowsum(K), K = exp(-d2/h2) * w.
// One wave handles 16 query rows; loops over all 8192 sources in 16-tiles.
__global__ void __launch_bounds__(256) ms_step(const float4* __restrict__ pq,
                                               const float4* __restrict__ yin,
                                               float4* __restrict__ yout, int n) {
  __shared__ float4 lds[256];
  const int tid  = threadIdx.x;
  const int lane = tid & 31;
  const int wave = tid >> 5;
  const int col  = lane & 15;      // query index within the 16-row tile
  const int hi   = lane >> 4;      // which 8-row half of the C matrix
  const int i    = (blockIdx.x * 8 + wave) * 16 + col;

  float4 yi = yin[i];
  const float y2C = yi.w * MS_CC;
  v2f B;                            // B(4x16): columns = queries {x,y | z,0}
  B.x = hi ? yi.z : yi.x;
  B.y = hi ? 0.0f : yi.y;

  float nx = 0.0f, ny = 0.0f, nz = 0.0f, den = 0.0f;

  for (int jc = 0; jc < n; jc += 256) {
    __syncthreads();
    lds[tid] = pq[jc + tid];
    __syncthreads();
    for (int jt = 0; jt < 256; jt += 16) {
      float4 aq = lds[jt + col];   // A(16x4): rows = sources {x,y | z,0}
      v2f A;
      A.x = hi ? aq.z : aq.x;
      A.y = hi ? 0.0f : aq.y;
      v8f S = wmma_dot(A, B);      // S[r] = q_{jt+hi*8+r} . y_i
      const int rb = jt + (hi << 3);
      #pragma unroll
      for (int r = 0; r < 8; ++r) {
        float4 p = lds[rb + r];
        // K = exp2(2C*S + log2(w_j) - C*|q_j|^2 - C*|y_i|^2) = exp(-d2/h2)*w_j
        float K = fast_exp2(fmaf(S[r], MS_2C, p.w - y2C));
        nx = fmaf(K, p.x, nx);
        ny = fmaf(K, p.y, ny);
        nz = fmaf(K, p.z, nz);
        den += K;
      }
    }
  }
  // merge the two 8-row halves (lanes L and L+16 hold same column i)
  nx  += __shfl_xor(nx, 16, 32);
  ny  += __shfl_xor(ny, 16, 32);
  nz  += __shfl_xor(nz, 16, 32);
  den += __shfl_xor(den, 16, 32);
  if (hi == 0) {
    float inv = 1.0f / den;
    float ox = nx * inv, oy = ny * inv, oz = nz * inv;
    yout[i] = make_float4(ox, oy, oz, fmaf(ox, ox, fmaf(oy, oy, oz * oz)));
  }
}

// packedY[j] = {yx, yy, yz, log2(w_j) - |y_j|^2 * C}
__global__ void __launch_bounds__(256) ms_ypack(const float4* __restrict__ y,
                                                const float* __restrict__ attn,
                                                float4* __restrict__ py, int n) {
  int j = blockIdx.x * blockDim.x + threadIdx.x;
  if (j >= n) return;
  float4 v = y[j];
  py[j] = make_float4(v.x, v.y, v.z, __log2f(attn[j]) - v.w * MS_CC);
}

// density_i = sum_j exp(-|y_i - y_j|^2 / h2) * w_j
__global__ void __launch_bounds__(256) ms_density(const float4* __restrict__ py,
                                                  const float4* __restrict__ yin,
                                                  float* __restrict__ dens, int n) {
  __shared__ float4 lds[256];
  const int tid  = threadIdx.x;
  const int lane = tid & 31;
  const int wave = tid >> 5;
  const int col  = lane & 15;
  const int hi   = lane >> 4;
  const int i    = (blockIdx.x * 8 + wave) * 16 + col;

  float4 yi = yin[i];
  const float y2C = yi.w * MS_CC;
  v2f B;
  B.x = hi ? yi.z : yi.x;
  B.y = hi ? 0.0f : yi.y;

  float den = 0.0f;
  for (int jc = 0; jc < n; jc += 256) {
    __syncthreads();
    lds[tid] = py[jc + tid];
    __syncthreads();
    for (int jt = 0; jt < 256; jt += 16) {
      float4 aq = lds[jt + col];
      v2f A;
      A.x = hi ? aq.z : aq.x;
      A.y = hi ? 0.0f : aq.y;
      v8f S = wmma_dot(A, B);
      const int rb = jt + (hi << 3);
      #pragma unroll
      for (int r = 0; r < 8; ++r) {
        float4 p = lds[rb + r];
        den += fast_exp2(fmaf(S[r], MS_2C, p.w - y2C));
      }
    }
  }
  den += __shfl_xor(den, 16, 32);
  if (hi == 0) dens[i] = den;
}

// pa[j] = {0.5*(|y_j|^2 - h2), density_j}
__global__ void __launch_bounds__(256) ms_argpack(const float4* __restrict__ y,
                                                  const float* __restrict__ dens,
                                                  float2* __restrict__ pa, int n) {
  int j = blockIdx.x * blockDim.x + threadIdx.x;
  if (j >= n) return;
  pa[j] = make_float2(0.5f * (y[j].w - MS_H2), dens[j]);
}

// out[i] = y[argmax_j (within ? density_j : -inf)], within: d2(y_i,y_j)<=h2
// d2 <= h2  <=>  S - 0.5*|y_i|^2 >= 0.5*(|y_j|^2 - h2)
__global__ void __launch_bounds__(256) ms_argmax(const float4* __restrict__ y,
                                                 const float2* __restrict__ pa,
                                                 float* __restrict__ out, int n) {
  __shared__ float4 ldsY[256];
  __shared__ float2 ldsA[256];
  const int tid  = threadIdx.x;
  const int lane = tid & 31;
  const int wave = tid >> 5;
  const int col  = lane & 15;
  const int hi   = lane >> 4;
  const int i    = (blockIdx.x * 8 + wave) * 16 + col;

  float4 yi = y[i];
  const float h2i = 0.5f * yi.w;
  v2f B;
  B.x = hi ? yi.z : yi.x;
  B.y = hi ? 0.0f : yi.y;

  float bestd = -INFINITY;
  int   bestj = i;

  for (int jc = 0; jc < n; jc += 256) {
    __syncthreads();
    ldsY[tid] = y[jc + tid];
    ldsA[tid] = pa[jc + tid];
    __syncthreads();
    for (int jt = 0; jt < 256; jt += 16) {
      float4 aq = ldsY[jt + col];
      v2f A;
      A.x = hi ? aq.z : aq.x;
      A.y = hi ? 0.0f : aq.y;
      v8f S = wmma_dot(A, B);
      const int rb = jt + (hi << 3);
      #pragma unroll
      for (int r = 0; r < 8; ++r) {
        float2 a = ldsA[rb + r];
        bool upd = ((S[r] - h2i) >= a.x) && (a.y > bestd);  // strict > keeps first
        bestd = upd ? a.y : bestd;
        bestj = upd ? (jc + rb + r) : bestj;
      }
    }
  }
  // merge halves: higher density wins; tie -> smaller index (first occurrence)
  float od = __shfl_xor(bestd, 16, 32);
  int   oj = __shfl_xor(bestj, 16, 32);
  bool take = (od > bestd) || ((od == bestd) && (oj < bestj));
  bestd = take ? od : bestd;
  bestj = take ? oj : bestj;
  if (hi == 0) {
    float4 m = y[bestj];
    out[3 * i + 0] = m.x;
    out[3 * i + 1] = m.y;
    out[3 * i + 2] = m.z;
  }
}

extern "C" void kernel_launch(void* const* d_in, const int* in_sizes, int n_in,
                              void* d_out, int out_size, void* d_ws, size_t ws_size,
                              hipStream_t stream) {
  (void)n_in; (void)out_size; (void)ws_size;
  const float* q    = (const float*)d_in[0];
  const float* attn = (const float*)d_in[1];
  const int n = in_sizes[1];            // 8192 points

  // workspace layout (float4-aligned):
  float4* pq   = (float4*)d_ws;         // n * 16 B
  float4* ya   = pq + n;                // n * 16 B
  float4* yb   = ya + n;                // n * 16 B
  float4* py   = yb + n;                // n * 16 B
  float*  dens = (float*)(py + n);      // n * 4 B
  float2* pa   = (float2*)(dens + n);   // n * 8 B

  dim3 blk(256);
  dim3 gpack((n + 255) / 256);          // 1 point / thread
  dim3 gtile(n / 128);                  // 16 rows / wave, 8 waves / block

  ms_pack<<<gpack, blk, 0, stream>>>(q, attn, pq, ya, n);

  float4* cur = ya;
  float4* nxt = yb;
  for (int it = 0; it < 10; ++it) {
    ms_step<<<gtile, blk, 0, stream>>>(pq, cur, nxt, n);
    float4* t = cur; cur = nxt; nxt = t;
  }
  // 10 iterations -> final y is back in `ya` (== cur)

  ms_ypack<<<gpack, blk, 0, stream>>>(cur, attn, py, n);
  ms_density<<<gtile, blk, 0, stream>>>(py, cur, dens, n);
  ms_argpack<<<gpack, blk, 0, stream>>>(cur, dens, pa, n);
  ms_argmax<<<gtile, blk, 0, stream>>>(cur, pa, (float*)d_out, n);
}